// AdaptiveQuerySelector_68247030333455
// MI455X (gfx1250) — compile-verified
//
#include <hip/hip_runtime.h>
#include <math.h>

// ---------------------------------------------------------------------------
// AdaptiveQuerySelector for MI455X (gfx1250, wave32)
//
// D=1024, N=16384, MAX_NEIGHBORS=10.
// Memory-bound (~10.5 MB traffic vs 42 MFLOP) -> keep fp32 end to end and use
// V_WMMA_F32_16X16X4_F32 for the (10x2048)@(2048x1024) attention GEMM so the
// matrix pipes do the math at reference precision while W_att1 is streamed
// from HBM exactly once. Split-K across 4 waves/block for 4x memory-level
// parallelism and 4x shorter WMMA accumulation chains.
// ---------------------------------------------------------------------------

typedef float v2f __attribute__((ext_vector_type(2)));
typedef float v8f __attribute__((ext_vector_type(8)));

#define D_DIM 1024
#define N_DIM 16384
#define MAXK  10
#define KWAVES 4                 // waves per block in the GEMM (split-K)
#define KSLICE (2 * D_DIM / KWAVES)   // 512 K-values per wave

// Workspace layout:
//   floats[0..15]  : score accumulators (zeroed each launch by count kernel)
//   ints at +64B   : meta[0] = dynamic_k, meta[1..10] = neighbor indices

// ---------------------------------------------------------------------------
// Kernel 1: neighbor-count MLP  (Linear 1024->512, ReLU, Linear 512->1, sigmoid)
// ---------------------------------------------------------------------------
__global__ void aqs_count_kernel(const float* __restrict__ tgt,
                                 const float* __restrict__ Wc1,   // [1024,512]
                                 const float* __restrict__ bc1,   // [512]
                                 const float* __restrict__ Wc2,   // [512]
                                 const float* __restrict__ bc2,   // [1]
                                 float* __restrict__ scores,
                                 int* __restrict__ meta) {
  __shared__ float red[512];
  const int j = threadIdx.x;            // output column of Wc1, 0..511
  float acc = 0.f;
  for (int i = 0; i < D_DIM; ++i)       // coalesced across j
    acc = fmaf(tgt[i], Wc1[i * 512 + j], acc);
  const float h = fmaxf(acc + bc1[j], 0.f);
  red[j] = h * Wc2[j];
  __syncthreads();
  for (int s = 256; s > 0; s >>= 1) {
    if (j < s) red[j] += red[j + s];
    __syncthreads();
  }
  if (j == 0) {
    const float z    = red[0] + bc2[0];
    const float frac = 1.f / (1.f + expf(-z));
    int k = (int)floorf(frac * 10.f);
    k = k < 1 ? 1 : (k > MAXK ? MAXK : k);
    meta[0] = k;
  }
  if (j < 16) scores[j] = 0.f;          // reset atomics target every launch
}

// ---------------------------------------------------------------------------
// Kernel 2: top-11 over the target's similarity row + stable compaction
// (tie-break on smaller index matches jax.lax.top_k stability)
// ---------------------------------------------------------------------------
__global__ void aqs_topk_kernel(const float* __restrict__ sim,
                                const int* __restrict__ tgt_idx,
                                int* __restrict__ meta) {
  __shared__ float sval[1024];
  __shared__ int   sidx[1024];
  __shared__ int   sel[MAXK + 1];
  const int t   = threadIdx.x;
  const int tgt = *tgt_idx;
  const float* row = sim + (size_t)tgt * N_DIM;

  float v[16];
  #pragma unroll
  for (int i = 0; i < 16; ++i) v[i] = row[i * 1024 + t];   // coalesced
  unsigned taken = 0;

  for (int p = 0; p < MAXK + 1; ++p) {
    float best = -INFINITY;
    int   bidx = 0x7FFFFFFF;
    #pragma unroll
    for (int i = 0; i < 16; ++i) {
      if (taken & (1u << i)) continue;
      const float x = v[i];
      const int   e = i * 1024 + t;
      if (x > best || (x == best && e < bidx)) { best = x; bidx = e; }
    }
    sval[t] = best; sidx[t] = bidx;
    __syncthreads();
    for (int s = 512; s > 0; s >>= 1) {
      if (t < s) {
        const float a = sval[t], b = sval[t + s];
        const int  ai = sidx[t], bi = sidx[t + s];
        if (b > a || (b == a && bi < ai)) { sval[t] = b; sidx[t] = bi; }
      }
      __syncthreads();
    }
    const int win = sidx[0];
    if (t == 0) sel[p] = win;
    if ((win & 1023) == t) taken |= 1u << (win >> 10);     // owner marks taken
    __syncthreads();
  }

  if (t == 0) {
    int c = 0;
    for (int p = 0; p < MAXK + 1 && c < MAXK; ++p)
      if (sel[p] != tgt) meta[1 + c++] = sel[p];           // stable compaction
  }
}

// ---------------------------------------------------------------------------
// Kernel 3: attention GEMM via V_WMMA_F32_16X16X4_F32, split-K over 4 waves.
// 64 blocks x 128 threads; each block owns a 16-column tile of N=1024, each
// wave owns a 512-wide K slice (slice boundaries align with the
// target/neighbor halves of `pair`, so A has one uniform base per wave).
// Partial 16x16 tiles reduced through LDS, then fused bias + ReLU +
// (.@ W_att2) row-reduction -> atomic partial scores.
//
// f32 16x16x4 fragment layout (ISA 7.12.2):
//   A: lane = 16*half + m holds A[m, k0+2*half + {0,1}] in VGPR {0,1}
//   B: lane = 16*half + n holds B[k0+2*half + {0,1}, n] in VGPR {0,1}
//   C: lane = 16*half + n holds D[v + 8*half, n] in VGPR v (v=0..7)
// ---------------------------------------------------------------------------
__global__ void aqs_attn_kernel(const float* __restrict__ tgt,
                                const float* __restrict__ emb,  // [N,D]
                                const float* __restrict__ W1,   // [2048,1024]
                                const float* __restrict__ b1,   // [1024]
                                const float* __restrict__ W2,   // [1024]
                                const int*   __restrict__ meta,
                                float* __restrict__ scores) {
  __shared__ float cred[KWAVES][16][16];

  const int tid  = threadIdx.x;        // 0..127
  const int wave = tid >> 5;           // 0..3 : K-slice owner
  const int lane = tid & 31;
  const int half = lane >> 4;          // K sub-pair select
  const int l    = lane & 15;          // A row m / B,D column-in-tile
  const int n    = blockIdx.x * 16 + l;

  // K range for this wave: [wave*512, wave*512+512). Waves 0,1 read the
  // target half of `pair`; waves 2,3 read the gathered neighbor rows.
  const int kbase = wave * KSLICE;
  // Neighbor row pointer for A row m=l; padding rows (m>=10) read a dummy
  // row — their D rows are never consumed.
  const float* nbr  = emb + (size_t)meta[1 + (l < MAXK ? l : 0)] * D_DIM;
  const float* Asrc = (kbase < D_DIM) ? (tgt + kbase) : (nbr + (kbase - D_DIM));
  const float* Ap   = Asrc + 2 * half;
  const float* Bp   = W1 + (size_t)(kbase + 2 * half) * D_DIM + n;

  v8f c = {0.f, 0.f, 0.f, 0.f, 0.f, 0.f, 0.f, 0.f};
  for (int k0 = 0; k0 < KSLICE; k0 += 4) {
    v2f a; a.x = Ap[k0];                    a.y = Ap[k0 + 1];
    v2f b; b.x = Bp[(size_t)k0 * D_DIM];    b.y = Bp[(size_t)(k0 + 1) * D_DIM];
    c = __builtin_amdgcn_wmma_f32_16x16x4_f32(
            /*neg_a=*/false, a, /*neg_b=*/false, b,
            /*c_mod=*/(short)0, c, /*reuse_a=*/false, /*reuse_b=*/false);
  }

  // ---- cross-wave split-K reduction through LDS --------------------------
  #pragma unroll
  for (int v = 0; v < 8; ++v)
    cred[wave][v + half * 8][l] = c[v];
  __syncthreads();

  if (wave == 0) {
    // ---- fused epilogue: +bias, ReLU, *W_att2, reduce over 16 columns ----
    const float bias = b1[n];
    const float w2   = W2[n];
    #pragma unroll
    for (int v = 0; v < 8; ++v) {
      const int mr = v + half * 8;                     // D row of this element
      float acc = cred[0][mr][l];
      #pragma unroll
      for (int w = 1; w < KWAVES; ++w) acc += cred[w][mr][l];
      float val = fmaxf(acc + bias, 0.f) * w2;
      for (int off = 8; off >= 1; off >>= 1)           // sum over the 16 lanes
        val += __shfl_xor(val, off, 16);               //   holding this row
      if (l == 0 && mr < MAXK)
        atomicAdd(&scores[mr], val);
    }
  }
}

// ---------------------------------------------------------------------------
// Kernel 4: masked softmax + weighted aggregation + output packing
// ---------------------------------------------------------------------------
__global__ void aqs_final_kernel(const float* __restrict__ emb,
                                 const float* __restrict__ b2,
                                 const int*   __restrict__ meta,
                                 const float* __restrict__ scores,
                                 float* __restrict__ out) {
  __shared__ float w[MAXK];
  __shared__ int   nid[MAXK];
  const int t = threadIdx.x;
  if (t == 0) {
    const int k = meta[0];
    float s[MAXK];
    float mx = -INFINITY;
    for (int m = 0; m < MAXK; ++m) {
      const float v = (m < k) ? (scores[m] + b2[0]) : -INFINITY;
      s[m] = v;
      mx = fmaxf(mx, v);
    }
    float sum = 0.f;
    for (int m = 0; m < MAXK; ++m) {
      const float e = (s[m] == -INFINITY) ? 0.f : expf(s[m] - mx);
      w[m] = e; sum += e;
    }
    for (int m = 0; m < MAXK; ++m) {
      w[m] /= sum;
      nid[m] = meta[1 + m];
      out[D_DIM + m]             = w[m];           // weights
      out[D_DIM + MAXK + m]      = (float)nid[m];  // neighbor_idx
      out[D_DIM + 2 * MAXK + m]  = w[m];           // weights (again)
    }
  }
  __syncthreads();
  for (int d = t; d < D_DIM; d += blockDim.x) {
    float a = 0.f;
    for (int m = 0; m < MAXK; ++m)
      a = fmaf(w[m], emb[(size_t)nid[m] * D_DIM + d], a);
    out[d] = a;                                    // aggregated
  }
}

// ---------------------------------------------------------------------------
extern "C" void kernel_launch(void* const* d_in, const int* in_sizes, int n_in,
                              void* d_out, int out_size, void* d_ws, size_t ws_size,
                              hipStream_t stream) {
  const float* tgt = (const float*)d_in[0];   // [1024]
  const float* emb = (const float*)d_in[1];   // [16384,1024]
  const float* sim = (const float*)d_in[2];   // [16384,16384]
  const float* W1  = (const float*)d_in[3];   // [2048,1024]
  const float* b1  = (const float*)d_in[4];   // [1024]
  const float* W2  = (const float*)d_in[5];   // [1024,1]
  const float* b2  = (const float*)d_in[6];   // [1]
  const float* Wc1 = (const float*)d_in[7];   // [1024,512]
  const float* bc1 = (const float*)d_in[8];   // [512]
  const float* Wc2 = (const float*)d_in[9];   // [512,1]
  const float* bc2 = (const float*)d_in[10];  // [1]
  const int*   tix = (const int*)d_in[11];    // [1]

  float* scores = (float*)d_ws;
  int*   meta   = (int*)((char*)d_ws + 64);
  float* out    = (float*)d_out;              // 1024 + 10 + 10 + 10 = 1054

  aqs_count_kernel<<<1, 512, 0, stream>>>(tgt, Wc1, bc1, Wc2, bc2, scores, meta);
  aqs_topk_kernel <<<1, 1024, 0, stream>>>(sim, tix, meta);
  aqs_attn_kernel <<<D_DIM / 16, 32 * KWAVES, 0, stream>>>(tgt, emb, W1, b1, W2, meta, scores);
  aqs_final_kernel<<<1, 256, 0, stream>>>(emb, b2, meta, scores, out);
}